// VariationalGraphAutoEncoder_39161511805133
// MI455X (gfx1250) — compile-verified
//
#include <hip/hip_runtime.h>
#include <hip/hip_bf16.h>
#include <math.h>

typedef float v2f __attribute__((ext_vector_type(2)));
typedef float v8f __attribute__((ext_vector_type(8)));

// ------------------------------------------------------------------
// degree count + dinv = rsqrt(deg + 1)
// ------------------------------------------------------------------
__global__ void deg_count_kernel(const int* __restrict__ ei, float* __restrict__ deg, int E) {
    int e = blockIdx.x * blockDim.x + threadIdx.x;
    if (e < E) {
        int d = ei[(size_t)E + e];           // row 1 = dst
        unsafeAtomicAdd(&deg[d], 1.0f);
    }
}

__global__ void dinv_kernel(float* __restrict__ deg, int N) {
    int i = blockIdx.x * blockDim.x + threadIdx.x;
    if (i < N) deg[i] = rsqrtf(deg[i] + 1.0f);
}

// ------------------------------------------------------------------
// WMMA f32 GEMM:  Y[nrows,NOUT] = X[nrows,K] @ W[K,NOUT]
// 256 threads = 8 waves; W staged in LDS as WMMA-ready float2 pairs;
// each wave: 16 rows, all NOUT/16 col tiles kept in registers.
// ------------------------------------------------------------------
template <int K, int NOUT>
__global__ __launch_bounds__(256) void gemm_wmma_kernel(
    const float* __restrict__ X, const float* __restrict__ W,
    float* __restrict__ Y, int nrows)
{
    constexpr int NT = NOUT / 16;
    // sW[(k/2)*NOUT + n] = { W[k][n], W[k+1][n] }  (k even) -> one ds_load_b64
    // per B fragment, already in an even-aligned VGPR pair for WMMA.
    __shared__ float2 sW[(K / 2) * NOUT];
    for (int i = threadIdx.x; i < (K / 2) * NOUT; i += 256) {
        int kh = i / NOUT;            // NOUT is pow2 -> shift
        int n  = i & (NOUT - 1);
        sW[i] = make_float2(W[(size_t)(2 * kh) * NOUT + n],
                            W[(size_t)(2 * kh + 1) * NOUT + n]);
    }
    __syncthreads();

    const int wave = threadIdx.x >> 5;
    const int lane = threadIdx.x & 31;
    const int half = lane >> 4;      // lanes 16-31
    const int lm   = lane & 15;
    const int row0 = (blockIdx.x * 8 + wave) * 16;
    if (row0 >= nrows) return;       // wave-uniform: whole tile out of range

    // Clamp the source row once (branchless). Garbage in an out-of-range A
    // row only affects the same out-of-range D row, which is never stored.
    const int arow = row0 + lm;
    const float* xrow = X + (size_t)(arow < nrows ? arow : nrows - 1) * K;

    v8f acc[NT] = {};

    for (int k = 0; k < K; k += 4) {
        // A 16x4 fragment: VGPR v holds K = k + 2*half + v for row M = lm
        const float2 t = *(const float2*)(xrow + k + 2 * half);
        v2f a; a.x = t.x; a.y = t.y;
        const float2* wrow = sW + ((k + 2 * half) >> 1) * NOUT;
#pragma unroll
        for (int ct = 0; ct < NT; ++ct) {
            // B 4x16 fragment: single b64 LDS read, col N = ct*16 + lm
            const float2 u = wrow[ct * 16 + lm];
            v2f b; b.x = u.x; b.y = u.y;
            acc[ct] = __builtin_amdgcn_wmma_f32_16x16x4_f32(
                false, a, false, b, (short)0, acc[ct], false, false);
        }
    }

    // D layout: VGPR v = row (v + 8*half), col = ct*16 + lm
    if (row0 + 16 <= nrows) {        // full tile: unguarded stores
#pragma unroll
        for (int ct = 0; ct < NT; ++ct)
#pragma unroll
            for (int v = 0; v < 8; ++v)
                Y[(size_t)(row0 + v + 8 * half) * NOUT + ct * 16 + lm] = acc[ct][v];
    } else {
#pragma unroll
        for (int ct = 0; ct < NT; ++ct)
#pragma unroll
            for (int v = 0; v < 8; ++v) {
                int r = row0 + v + 8 * half;
                if (r < nrows) Y[(size_t)r * NOUT + ct * 16 + lm] = acc[ct][v];
            }
    }
}

// ------------------------------------------------------------------
// dest = lin * dinv^2 + bias   (self-loop term + bias; overwrites dest)
// ------------------------------------------------------------------
template <int NOUT>
__global__ void selfloop_bias_kernel(const float* __restrict__ lin,
                                     const float* __restrict__ dinv,
                                     const float* __restrict__ bias,
                                     float* __restrict__ dest, int nrows)
{
    constexpr int SH = (NOUT == 128) ? 7 : 6;
    long long idx = (long long)blockIdx.x * blockDim.x + threadIdx.x;
    if (idx >= (long long)nrows * NOUT) return;
    int r = (int)(idx >> SH);
    int c = (int)(idx & (NOUT - 1));
    float di = dinv[r];
    dest[idx] = lin[idx] * di * di + bias[c];
}

// ------------------------------------------------------------------
// edge scatter: dest[dst] += lin[src] * dinv[src]*dinv[dst]
// 32 (or 16) consecutive lanes handle one edge's row in float4 chunks.
// ------------------------------------------------------------------
template <int NOUT>
__global__ void scatter_kernel(const float* __restrict__ lin,
                               const int* __restrict__ ei,
                               const float* __restrict__ dinv,
                               float* __restrict__ dest, int E)
{
    constexpr int G  = NOUT / 4;
    constexpr int SH = (G == 32) ? 5 : 4;
    long long gid = (long long)blockIdx.x * blockDim.x + threadIdx.x;
    if (gid >= (long long)E * G) return;
    int e = (int)(gid >> SH);
    int g = (int)(gid & (G - 1));
    int s = ei[e];
    int d = ei[(size_t)E + e];
    float nrm = dinv[s] * dinv[d];
    const float4 v = *(const float4*)(lin + (size_t)s * NOUT + g * 4);
    float* dp = dest + (size_t)d * NOUT + g * 4;
    unsafeAtomicAdd(dp + 0, v.x * nrm);
    unsafeAtomicAdd(dp + 1, v.y * nrm);
    unsafeAtomicAdd(dp + 2, v.z * nrm);
    unsafeAtomicAdd(dp + 3, v.w * nrm);
}

// ------------------------------------------------------------------
// activation: 1 = relu, 2 = sigmoid
// ------------------------------------------------------------------
__global__ void act_kernel(float* __restrict__ buf, long long n, int mode) {
    long long i = (long long)blockIdx.x * blockDim.x + threadIdx.x;
    if (i >= n) return;
    float v = buf[i];
    if (mode == 1)      v = fmaxf(v, 0.0f);
    else if (mode == 2) v = 1.0f / (1.0f + __expf(-v));
    buf[i] = v;
}

// z = mu + si * eps
__global__ void z_kernel(const float* __restrict__ mu, const float* __restrict__ si,
                         const float* __restrict__ eps, float* __restrict__ z, long long n) {
    long long i = (long long)blockIdx.x * blockDim.x + threadIdx.x;
    if (i < n) z[i] = fmaf(si[i], eps[i], mu[i]);
}

// ------------------------------------------------------------------
// host-side helpers
// ------------------------------------------------------------------
static void run_gemm(hipStream_t s, const float* X, const float* W, float* Y,
                     int n, int K, int NOUT) {
    int blocks = (n + 127) / 128;
    if (K == 64 && NOUT == 128)
        gemm_wmma_kernel<64, 128><<<blocks, 256, 0, s>>>(X, W, Y, n);
    else if (K == 128 && NOUT == 128)
        gemm_wmma_kernel<128, 128><<<blocks, 256, 0, s>>>(X, W, Y, n);
    else // 128 -> 64
        gemm_wmma_kernel<128, 64><<<blocks, 256, 0, s>>>(X, W, Y, n);
}

static void run_gcn_layer(hipStream_t s, const float* X, const float* W, const float* B,
                          const int* ei, const float* dinv,
                          float* lin, float* dest, int n, int E,
                          int K, int NOUT, int act) {
    run_gemm(s, X, W, lin, n, K, NOUT);
    long long tot = (long long)n * NOUT;
    int b1 = (int)((tot + 255) / 256);
    long long se = (long long)E * (NOUT / 4);
    int b2 = (int)((se + 255) / 256);
    if (NOUT == 128) {
        selfloop_bias_kernel<128><<<b1, 256, 0, s>>>(lin, dinv, B, dest, n);
        scatter_kernel<128><<<b2, 256, 0, s>>>(lin, ei, dinv, dest, E);
    } else {
        selfloop_bias_kernel<64><<<b1, 256, 0, s>>>(lin, dinv, B, dest, n);
        scatter_kernel<64><<<b2, 256, 0, s>>>(lin, ei, dinv, dest, E);
    }
    if (act) act_kernel<<<b1, 256, 0, s>>>(dest, tot, act);
}

extern "C" void kernel_launch(void* const* d_in, const int* in_sizes, int n_in,
                              void* d_out, int out_size, void* d_ws, size_t ws_size,
                              hipStream_t stream) {
    const int INF = 64, H = 128, ZD = 64, C = 2;
    const float* x   = (const float*)d_in[0];
    const float* eps = (const float*)d_in[1];
    const int*   ei  = (const int*)d_in[2];
    const float* w_in  = (const float*)d_in[3];  const float* b_in  = (const float*)d_in[4];
    const float* w_mu0 = (const float*)d_in[5];  const float* b_mu0 = (const float*)d_in[6];
    const float* w_mu  = (const float*)d_in[7];  const float* b_mu  = (const float*)d_in[8];
    const float* w_si0 = (const float*)d_in[9];  const float* b_si0 = (const float*)d_in[10];
    const float* w_si  = (const float*)d_in[11]; const float* b_si  = (const float*)d_in[12];
    const float* w_z   = (const float*)d_in[13]; const float* b_z   = (const float*)d_in[14];
    const float* w_d0  = (const float*)d_in[15]; const float* b_d0  = (const float*)d_in[16];
    const float* w_out = (const float*)d_in[17]; const float* b_out = (const float*)d_in[18];

    const int N = in_sizes[0] / INF;
    const int E = in_sizes[2] / 2;

    // d_out layout: logits [N*IN*C], mu [N*Z], si [N*Z]
    float* out    = (float*)d_out;
    float* logits = out;
    float* mu     = out + (size_t)N * INF * C;
    float* si     = mu + (size_t)N * ZD;

    // workspace: dinv | lin | bufh | bufa
    char* ws = (char*)d_ws;
    size_t off = 0;
    float* dinv = (float*)(ws + off); off += ((size_t)N * 4 + 255) & ~(size_t)255;
    float* lin  = (float*)(ws + off); off += ((size_t)N * H * 4 + 255) & ~(size_t)255;
    float* bufh = (float*)(ws + off); off += ((size_t)N * H * 4 + 255) & ~(size_t)255;
    float* bufa = (float*)(ws + off);

    // degree / dinv
    hipMemsetAsync(dinv, 0, (size_t)N * 4, stream);
    deg_count_kernel<<<(E + 255) / 256, 256, 0, stream>>>(ei, dinv, E);
    dinv_kernel<<<(N + 255) / 256, 256, 0, stream>>>(dinv, N);

    // encode
    run_gcn_layer(stream, x,    w_in,  b_in,  ei, dinv, lin, bufh, N, E, 64,  128, 1);
    run_gcn_layer(stream, bufh, w_mu0, b_mu0, ei, dinv, lin, bufa, N, E, 128, 128, 1);
    run_gcn_layer(stream, bufa, w_mu,  b_mu,  ei, dinv, lin, mu,   N, E, 128, 64,  1);
    run_gcn_layer(stream, bufh, w_si0, b_si0, ei, dinv, lin, bufa, N, E, 128, 128, 1);
    run_gcn_layer(stream, bufa, w_si,  b_si,  ei, dinv, lin, si,   N, E, 128, 64,  2);

    // reparameterize: z -> bufa
    long long nz = (long long)N * ZD;
    z_kernel<<<(int)((nz + 255) / 256), 256, 0, stream>>>(mu, si, eps, bufa, nz);

    // decode
    run_gcn_layer(stream, bufa, w_z,   b_z,   ei, dinv, lin, bufh,   N, E, 64,  128, 0);
    run_gcn_layer(stream, bufh, w_d0,  b_d0,  ei, dinv, lin, bufa,   N, E, 128, 128, 1);
    run_gcn_layer(stream, bufa, w_out, b_out, ei, dinv, lin, logits, N, E, 128, 128, 0);
}